// sinkhorn_19413252178438
// MI455X (gfx1250) — compile-verified
//
#include <hip/hip_runtime.h>
#include <math.h>

typedef __attribute__((ext_vector_type(2))) float v2f;
typedef __attribute__((ext_vector_type(8))) float v8f;

#define BATCH 32
#define MPTS  1000
#define NPTS  1000
#define DIM   128
#define N1    1001          // padded (dustbin) dimension
#define LDK   1024          // row stride of K in workspace
#define MU_INV 10.0f        // 1 / 0.1
#define EPSV   1e-12f

// ---------------------------------------------------------------------------
// row squared-norms: out[b*1024 + i] = sum_k x[b,i,k]^2
// ---------------------------------------------------------------------------
__global__ void norms_kernel(const float* __restrict__ x, float* __restrict__ out,
                             int rows) {
    int b = blockIdx.y;
    int i = blockIdx.x * blockDim.x + threadIdx.x;
    if (i >= rows) return;
    const float4* p = (const float4*)(x + ((size_t)b * rows + i) * DIM);
    float s = 0.f;
#pragma unroll
    for (int k = 0; k < DIM / 4; ++k) {
        float4 q = p[k];
        s += q.x * q.x + q.y * q.y + q.z * q.z + q.w * q.w;
    }
    out[b * 1024 + i] = s;
}

// ---------------------------------------------------------------------------
// marginals r, c and u0 = r   (i in [0,1023]; valid region [0,1000])
// ---------------------------------------------------------------------------
__global__ void marginals_kernel(const int* __restrict__ np2, const int* __restrict__ np3,
                                 float* __restrict__ rA, float* __restrict__ cA,
                                 float* __restrict__ uA) {
    int b = blockIdx.y;
    int i = blockIdx.x * blockDim.x + threadIdx.x;
    int n2p = np2[b], n3p = np3[b];
    float rv = (i <= n2p) ? 1.0f / ((float)n2p + 1.0f) : 0.0f;
    float cv = (i <= n3p) ? 1.0f / ((float)n3p + 1.0f) : 0.0f;
    rA[b * 1024 + i] = rv;
    cA[b * 1024 + i] = cv;
    uA[b * 1024 + i] = rv;
}

// ---------------------------------------------------------------------------
// Fused WMMA GEMM (x2 · x3^T) + distance + exp -> K (unnormalized)
// block = 256 threads = 8 waves arranged 2(m) x 4(n), tile = 32x64, K-dim 128
// ---------------------------------------------------------------------------
__global__ __launch_bounds__(256) void gemm_dist_kernel(
    const float* __restrict__ x2, const float* __restrict__ x3,
    const float* __restrict__ n2, const float* __restrict__ n3,
    float* __restrict__ K) {
    __shared__ float As[32][132];   // pad 132: bank-conflict-free column reads
    __shared__ float Bs[64][132];

    const int b  = blockIdx.z;
    const int m0 = blockIdx.y * 32;
    const int n0 = blockIdx.x * 64;
    const int tid = threadIdx.x;

    // stage A tile (32 x 128) and B tile (64 x 128), float4 coalesced
    {
        const float* srcA = x2 + (size_t)b * MPTS * DIM;
        for (int t = tid; t < 32 * 32; t += 256) {        // 1024 float4
            int row = t >> 5, c4 = t & 31;
            int gm  = m0 + row;
            float4 v = make_float4(0.f, 0.f, 0.f, 0.f);
            if (gm < MPTS) v = ((const float4*)(srcA + (size_t)gm * DIM))[c4];
            As[row][c4 * 4 + 0] = v.x; As[row][c4 * 4 + 1] = v.y;
            As[row][c4 * 4 + 2] = v.z; As[row][c4 * 4 + 3] = v.w;
        }
        const float* srcB = x3 + (size_t)b * NPTS * DIM;
        for (int t = tid; t < 64 * 32; t += 256) {        // 2048 float4
            int row = t >> 5, c4 = t & 31;
            int gn  = n0 + row;
            float4 v = make_float4(0.f, 0.f, 0.f, 0.f);
            if (gn < NPTS) v = ((const float4*)(srcB + (size_t)gn * DIM))[c4];
            Bs[row][c4 * 4 + 0] = v.x; Bs[row][c4 * 4 + 1] = v.y;
            Bs[row][c4 * 4 + 2] = v.z; Bs[row][c4 * 4 + 3] = v.w;
        }
    }
    __syncthreads();

    const int wave = tid >> 5, lane = tid & 31;
    const int wm = wave & 1, wn = wave >> 1;      // 2 x 4 wave grid
    const int half = lane >> 4, lrow = lane & 15; // WMMA lane decomposition

    // A frag: lane(0-15)=M, vgpr{0,1}=K{0,1}; lanes 16-31 -> K{2,3}
    // B frag: lane(0-15)=N, vgpr{0,1}=K{0,1}; lanes 16-31 -> K{2,3}
    //         B[k][n] = x3[n][k]  =>  same row-indexed load as A
    const float* arow = &As[wm * 16 + lrow][0];
    const float* brow = &Bs[wn * 16 + lrow][0];

    v8f acc = {};
#pragma unroll
    for (int k = 0; k < DIM; k += 4) {
        v2f a, bb;
        a[0]  = arow[k + 2 * half];
        a[1]  = arow[k + 2 * half + 1];
        bb[0] = brow[k + 2 * half];
        bb[1] = brow[k + 2 * half + 1];
        acc = __builtin_amdgcn_wmma_f32_16x16x4_f32(
            /*neg_a=*/false, a, /*neg_b=*/false, bb,
            /*c_mod=*/(short)0, acc, /*reuse_a=*/false, /*reuse_b=*/false);
    }

    // epilogue: D lane layout -> N = lrow, M = r + 8*half
    const int gn = n0 + wn * 16 + lrow;
    if (gn < NPTS) {
        const float n3v = n3[b * 1024 + gn];
        const int gmb   = m0 + wm * 16 + half * 8;
#pragma unroll
        for (int r = 0; r < 8; ++r) {
            int gm = gmb + r;
            if (gm < MPTS) {
                float g    = acc[r];
                float d2   = n2[b * 1024 + gm] + n3v - 2.0f * g;
                float dist = sqrtf(fmaxf(d2, 1e-30f));
                K[((size_t)b * N1 + gm) * LDK + gn] = __expf(-dist * MU_INV);
            }
        }
    }
}

// ---------------------------------------------------------------------------
// dustbin row/col: exp(0) = 1
// ---------------------------------------------------------------------------
__global__ void dustbin_kernel(float* __restrict__ K) {
    int b = blockIdx.y;
    int j = blockIdx.x * blockDim.x + threadIdx.x;
    if (j < N1)   K[((size_t)b * N1 + MPTS) * LDK + j]    = 1.0f;  // row 1000
    if (j < MPTS) K[((size_t)b * N1 + j)    * LDK + NPTS] = 1.0f;  // col 1000
}

// ---------------------------------------------------------------------------
// deterministic per-batch total sum of K (fixed-order tree, no atomics)
// ---------------------------------------------------------------------------
__global__ __launch_bounds__(256) void sum_kernel(const float* __restrict__ K,
                                                  float* __restrict__ S) {
    __shared__ float red[256];
    const int b = blockIdx.x, tid = threadIdx.x;
    float acc = 0.f;
    for (int i = 0; i < N1; ++i) {
        const float* row = K + ((size_t)b * N1 + i) * LDK;
        for (int j = tid; j < N1; j += 256) acc += row[j];
    }
    red[tid] = acc;
    __syncthreads();
    for (int s = 128; s > 0; s >>= 1) {
        if (tid < s) red[tid] += red[tid + s];
        __syncthreads();
    }
    if (tid == 0) S[b] = red[0];
}

// ---------------------------------------------------------------------------
// v = c / max((K^T u)/S, eps)   -- column-parallel, coalesced, u in LDS
// ---------------------------------------------------------------------------
__global__ __launch_bounds__(256) void ktu_kernel(
    const float* __restrict__ K, const float* __restrict__ uA,
    const float* __restrict__ cA, const float* __restrict__ S,
    float* __restrict__ vA) {
    __shared__ float uS[1024];
    const int b = blockIdx.y, tid = threadIdx.x;
    for (int t = tid; t < 1024; t += 256) uS[t] = (t < N1) ? uA[b * 1024 + t] : 0.f;
    __syncthreads();
    const int j = blockIdx.x * 256 + tid;
    if (j >= N1) return;
    const float* Kb = K + (size_t)b * N1 * LDK;
    float acc = 0.f;
    for (int i = 0; i < N1; ++i) acc += Kb[(size_t)i * LDK + j] * uS[i];
    float den = fmaxf(acc / S[b], EPSV);
    vA[b * 1024 + j] = cA[b * 1024 + j] / den;
}

// ---------------------------------------------------------------------------
// u = r / max((K v)/S, eps)   -- one wave32 per row, shfl reduction
// ---------------------------------------------------------------------------
__global__ __launch_bounds__(256) void kv_kernel(
    const float* __restrict__ K, const float* __restrict__ vA,
    const float* __restrict__ rA, const float* __restrict__ S,
    float* __restrict__ uA) {
    __shared__ float vS[1024];
    const int b = blockIdx.y, tid = threadIdx.x;
    for (int t = tid; t < 1024; t += 256) vS[t] = (t < N1) ? vA[b * 1024 + t] : 0.f;
    __syncthreads();
    const int wave = tid >> 5, lane = tid & 31;
    const int i = blockIdx.x * 8 + wave;
    if (i >= N1) return;
    const float* row = K + ((size_t)b * N1 + i) * LDK;
    float acc = 0.f;
    for (int j = lane; j < N1; j += 32) acc += row[j] * vS[j];
#pragma unroll
    for (int off = 16; off > 0; off >>= 1) acc += __shfl_xor(acc, off, 32);
    if (lane == 0) {
        float den = fmaxf(acc / S[b], EPSV);
        uA[b * 1024 + i] = rA[b * 1024 + i] / den;
    }
}

// ---------------------------------------------------------------------------
// P = u * (K/S) * v^T  -> packed (b, 1001, 1001)
// ---------------------------------------------------------------------------
__global__ void final_kernel(const float* __restrict__ K, const float* __restrict__ uA,
                             const float* __restrict__ vA, const float* __restrict__ S,
                             float* __restrict__ out) {
    const int b = blockIdx.y, i = blockIdx.x, tid = threadIdx.x;
    const float scale = uA[b * 1024 + i] / S[b];
    const float* row = K + ((size_t)b * N1 + i) * LDK;
    float* orow = out + ((size_t)b * N1 + i) * N1;
    for (int j = tid; j < N1; j += 256)
        orow[j] = scale * row[j] * vA[b * 1024 + j];
}

// ---------------------------------------------------------------------------
extern "C" void kernel_launch(void* const* d_in, const int* in_sizes, int n_in,
                              void* d_out, int out_size, void* d_ws, size_t ws_size,
                              hipStream_t stream) {
    const float* x2  = (const float*)d_in[0];
    const float* x3  = (const float*)d_in[1];
    const int*   np2 = (const int*)d_in[2];
    const int*   np3 = (const int*)d_in[3];

    float* wsf = (float*)d_ws;
    const size_t Kelems = (size_t)BATCH * N1 * LDK;
    float* K  = wsf;
    float* n2 = K  + Kelems;
    float* n3 = n2 + (size_t)BATCH * 1024;
    float* rA = n3 + (size_t)BATCH * 1024;
    float* cA = rA + (size_t)BATCH * 1024;
    float* uA = cA + (size_t)BATCH * 1024;
    float* vA = uA + (size_t)BATCH * 1024;
    float* S  = vA + (size_t)BATCH * 1024;

    norms_kernel<<<dim3(4, BATCH), 256, 0, stream>>>(x2, n2, MPTS);
    norms_kernel<<<dim3(4, BATCH), 256, 0, stream>>>(x3, n3, NPTS);
    marginals_kernel<<<dim3(4, BATCH), 256, 0, stream>>>(np2, np3, rA, cA, uA);

    gemm_dist_kernel<<<dim3(16, 32, BATCH), 256, 0, stream>>>(x2, x3, n2, n3, K);
    dustbin_kernel<<<dim3(4, BATCH), 256, 0, stream>>>(K);
    sum_kernel<<<BATCH, 256, 0, stream>>>(K, S);

    for (int it = 0; it < 21; ++it) {
        ktu_kernel<<<dim3(4, BATCH), 256, 0, stream>>>(K, uA, cA, S, vA);
        kv_kernel<<<dim3(126, BATCH), 256, 0, stream>>>(K, vA, rA, S, uA);
    }

    final_kernel<<<dim3(N1, BATCH), 256, 0, stream>>>(K, uA, vA, S, (float*)d_out);
}